// DyShiftMax_18270790877205
// MI455X (gfx1250) — compile-verified
//
#include <hip/hip_runtime.h>
#include <hip/hip_bf16.h>

typedef __attribute__((ext_vector_type(2))) float v2f;
typedef __attribute__((ext_vector_type(8))) float v8f;

#define NBATCH 32
#define CCH    256
#define HIDN   64      // C / R
#define TOUTN  1024    // J*K*C
#define HWPIX  3136    // 56*56
#define HWVEC  784     // HWPIX / 4
#define SHIFT  64      // C / G

// ---------------------------------------------------------------------------
// Kernel 1: global average pool  theta[n*C+c] = mean(x[n,c,:,:])
// One 256-thread block per (n,c) plane; coalesced strided reads; LDS reduce.
// ---------------------------------------------------------------------------
__global__ __launch_bounds__(256) void gap_kernel(const float* __restrict__ x,
                                                  float* __restrict__ theta) {
  const int plane = blockIdx.x;                   // n*C + c  (8192 planes)
  const float* __restrict__ p = x + (size_t)plane * HWPIX;
  float s = 0.f;
  for (int i = threadIdx.x; i < HWPIX; i += 256) s += p[i];
  __shared__ float red[256];
  red[threadIdx.x] = s;
  __syncthreads();
#pragma unroll
  for (int off = 128; off > 0; off >>= 1) {
    if ((int)threadIdx.x < off) red[threadIdx.x] += red[threadIdx.x + off];
    __syncthreads();
  }
  if (threadIdx.x == 0) theta[plane] = red[0] * (1.0f / (float)HWPIX);
}

// ---------------------------------------------------------------------------
// Kernel 2: gating MLP with fp32 WMMA (V_WMMA_F32_16X16X4_F32).
//   h = relu(theta @ w1^T + b1)            [32,64]
//   t = 2*sigmoid(h @ w2^T + b2) - 1       [32,1024]
//   coefs[n, c*4 + j*2 + k] = t * lambdas[j] + alphas[k]
// Single 512-thread block (16 wave32s).
// fp32 WMMA operand layout (ISA 7.12.2):
//   A 16x4:  lane L: row = L&15, holds K = 2*(L>>4) + {0,1}    (2 VGPRs)
//   B 4x16:  lane L: col = L&15, holds K = 2*(L>>4) + {0,1}    (2 VGPRs)
//   C/D 16x16: VGPR g: row = g + 8*(L>>4), col = L&15          (8 VGPRs)
// ---------------------------------------------------------------------------
__global__ __launch_bounds__(512, 1) void
mlp_kernel(const float* __restrict__ theta,
           const float* __restrict__ w1, const float* __restrict__ b1,
           const float* __restrict__ w2, const float* __restrict__ b2,
           const float* __restrict__ lambdas, const float* __restrict__ alphas,
           float* __restrict__ coefs) {
  __shared__ float h_s[NBATCH][HIDN + 1];         // +1 pad: conflict-free column reads

  const int lane  = threadIdx.x & 31;
  const int wave  = threadIdx.x >> 5;
  const int row16 = lane & 15;
  const int koff  = (lane >> 4) << 1;             // 0 or 2 (K sub-offset within step)
  const int rhalf = (lane >> 4) << 3;             // 0 or 8 (C/D row half)

  // ---- Stage 1: theta[32,256] x w1^T[256,64] -> h[32,64]; 8 tiles, waves 0..7
  if (wave < 8) {
    const int m  = wave >> 2;                     // row tile 0..1
    const int nt = wave & 3;                      // col tile 0..3
    const int gm = m * 16 + row16;                // A row (batch)
    const int gn = nt * 16 + row16;               // B col (hidden unit)
    v8f acc = {};
    for (int k0 = 0; k0 < CCH; k0 += 4) {
      v2f a, b;
      a.x = theta[gm * CCH + k0 + koff];
      a.y = theta[gm * CCH + k0 + koff + 1];
      b.x = w1[gn * CCH + k0 + koff];             // B[k][n] = w1[n*CCH + k]
      b.y = w1[gn * CCH + k0 + koff + 1];
      acc = __builtin_amdgcn_wmma_f32_16x16x4_f32(false, a, false, b,
                                                  (short)0, acc, false, false);
    }
    const int col = nt * 16 + row16;
    const float bias = b1[col];
#pragma unroll
    for (int g = 0; g < 8; ++g) {
      const int r = m * 16 + g + rhalf;
      const float v = acc[g] + bias;
      h_s[r][col] = v > 0.f ? v : 0.f;
    }
  }
  __syncthreads();

  const float lam0 = lambdas[0], lam1 = lambdas[1];
  const float al0  = alphas[0],  al1  = alphas[1];

  // ---- Stage 2: h[32,64] x w2^T[64,1024] -> t[32,1024]; 128 tiles, 8 per wave
  for (int t8 = 0; t8 < 8; ++t8) {
    const int tid = wave * 8 + t8;                // 0..127
    const int m   = tid & 1;                      // row tile 0..1
    const int nt  = tid >> 1;                     // col tile 0..63
    const int gm  = m * 16 + row16;
    const int gn  = nt * 16 + row16;
    v8f acc = {};
#pragma unroll
    for (int k0 = 0; k0 < HIDN; k0 += 4) {
      v2f a, b;
      a.x = h_s[gm][k0 + koff];
      a.y = h_s[gm][k0 + koff + 1];
      b.x = w2[gn * HIDN + k0 + koff];            // B[k][n] = w2[n*HIDN + k]
      b.y = w2[gn * HIDN + k0 + koff + 1];
      acc = __builtin_amdgcn_wmma_f32_16x16x4_f32(false, a, false, b,
                                                  (short)0, acc, false, false);
    }
    const int col = nt * 16 + row16;              // 0..1023 = c*4 + j*2 + k
    const float bias = b2[col];
    const float lam  = ((col >> 1) & 1) ? lam1 : lam0;
    const float alp  = (col & 1) ? al1 : al0;
#pragma unroll
    for (int g = 0; g < 8; ++g) {
      const int r = m * 16 + g + rhalf;           // batch row 0..31
      const float v = acc[g] + bias;
      const float t = 2.f / (1.f + __expf(-v)) - 1.f;
      coefs[(size_t)r * TOUTN + col] = t * lam + alp;
    }
  }
}

// ---------------------------------------------------------------------------
// Kernel 3: streaming elementwise pass (the HBM-bound bulk of the op).
//   out[n,c,p] = max_k ( x[n,c,p]*coef[j=0,k] + x[n,(c+64)%256,p]*coef[j=1,k] )
// float4 (b128) loads/stores; 4 scalar coefs broadcast per plane.
// ---------------------------------------------------------------------------
__global__ __launch_bounds__(256) void dysm_main(const float* __restrict__ x,
                                                 const float* __restrict__ coefs,
                                                 float* __restrict__ out) {
  const int plane = blockIdx.y;                   // n*C + c
  const int n  = plane >> 8;
  const int c  = plane & (CCH - 1);
  const int c1 = (c + SHIFT) & (CCH - 1);         // roll(x,-s)[c] = x[(c+s)%C]
  const float4* __restrict__ p0 =
      (const float4*)(x + ((size_t)n * CCH + c) * HWPIX);
  const float4* __restrict__ p1 =
      (const float4*)(x + ((size_t)n * CCH + c1) * HWPIX);
  const float* __restrict__ cf = coefs + (size_t)plane * 4;
  const float c00 = cf[0];                        // j=0,k=0
  const float c01 = cf[1];                        // j=0,k=1
  const float c10 = cf[2];                        // j=1,k=0
  const float c11 = cf[3];                        // j=1,k=1

  const int i = blockIdx.x * 256 + threadIdx.x;   // vec4 index within plane
  if (i < HWVEC) {
    const float4 a = p0[i];
    const float4 b = p1[i];
    float4 r;
    r.x = fmaxf(fmaf(a.x, c00, b.x * c10), fmaf(a.x, c01, b.x * c11));
    r.y = fmaxf(fmaf(a.y, c00, b.y * c10), fmaf(a.y, c01, b.y * c11));
    r.z = fmaxf(fmaf(a.z, c00, b.z * c10), fmaf(a.z, c01, b.z * c11));
    r.w = fmaxf(fmaf(a.w, c00, b.w * c10), fmaf(a.w, c01, b.w * c11));
    ((float4*)(out + (size_t)plane * HWPIX))[i] = r;
  }
}

// ---------------------------------------------------------------------------
extern "C" void kernel_launch(void* const* d_in, const int* in_sizes, int n_in,
                              void* d_out, int out_size, void* d_ws, size_t ws_size,
                              hipStream_t stream) {
  const float* x       = (const float*)d_in[0];
  const float* w1      = (const float*)d_in[1];
  const float* b1      = (const float*)d_in[2];
  const float* w2      = (const float*)d_in[3];
  const float* b2      = (const float*)d_in[4];
  const float* lambdas = (const float*)d_in[5];
  const float* alphas  = (const float*)d_in[6];
  float* out = (float*)d_out;

  float* theta = (float*)d_ws;                    // 32*256  floats (32 KB)
  float* coefs = theta + NBATCH * CCH;            // 32*1024 floats (128 KB)

  gap_kernel<<<dim3(NBATCH * CCH), dim3(256), 0, stream>>>(x, theta);
  mlp_kernel<<<dim3(1), dim3(512), 0, stream>>>(theta, w1, b1, w2, b2,
                                                lambdas, alphas, coefs);
  dysm_main<<<dim3((HWVEC + 255) / 256, NBATCH * CCH), dim3(256), 0, stream>>>(
      x, coefs, out);
}